// Former2Mobile_25520695673124
// MI455X (gfx1250) — compile-verified
//
#include <hip/hip_runtime.h>
#include <hip/hip_bf16.h>

typedef __attribute__((ext_vector_type(2))) float v2f;
typedef __attribute__((ext_vector_type(8))) float v8f;

#define B_   32
#define C_   64
#define HW_  3136      // 56*56
#define M_   6
#define DIM_ 192
#define NH_  48        // heads(8) * M(6)
#define INNER_ 512

__device__ __forceinline__ v8f wmma_f32(v2f a, v2f b, v8f c) {
    // D = A(16x4 f32) * B(4x16 f32) + C(16x16 f32)
    return __builtin_amdgcn_wmma_f32_16x16x4_f32(false, a, false, b, (short)0, c, false, false);
}

// ---------------------------------------------------------------------------
// Kernel 1 (per batch): Kmat[b][r=h*6+m][c] = scale*(z[b,m,:]@Wk[:,h*64+c] + bk)
//                       Vo[b][r][c]        = (z[b,m,:]@Wv + bv)[h*64+:] @ Wo[h*64+:, c]
// Tiny (~100 MFLOP total), scalar VALU is fine.
// ---------------------------------------------------------------------------
__global__ __launch_bounds__(256)
void build_kvo_kernel(const float* __restrict__ z,  const float* __restrict__ Wk,
                      const float* __restrict__ bk, const float* __restrict__ Wv,
                      const float* __restrict__ bv, const float* __restrict__ Wo,
                      float* __restrict__ kws, float* __restrict__ vows)
{
    const int b = blockIdx.x;
    __shared__ float zs[M_ * DIM_];      // 4.6 KB
    __shared__ float vs[M_ * INNER_];    // 12.3 KB
    const float* zb = z + (size_t)b * (M_ * DIM_);
    for (int i = threadIdx.x; i < M_ * DIM_; i += 256) zs[i] = zb[i];
    __syncthreads();

    const float scale = 0.125f;  // 64^-0.5, folded into Kmat
    for (int i = threadIdx.x; i < M_ * INNER_; i += 256) {
        const int m = i / INNER_, col = i % INNER_;
        float accK = bk[col], accV = bv[col];
        const float* zr = zs + m * DIM_;
        for (int d = 0; d < DIM_; ++d) {
            const float zv = zr[d];
            accK = fmaf(zv, Wk[(size_t)d * INNER_ + col], accK);
            accV = fmaf(zv, Wv[(size_t)d * INNER_ + col], accV);
        }
        const int h = col >> 6, c = col & 63;
        kws[(size_t)b * (NH_ * C_) + (h * M_ + m) * C_ + c] = accK * scale;
        vs[i] = accV;
    }
    __syncthreads();

    for (int i = threadIdx.x; i < NH_ * C_; i += 256) {
        const int r = i / C_, c = i % C_;
        const int h = r / M_, m = r % M_;
        const float* vrow = vs + m * INNER_ + h * C_;       // v[b,h,m,cc]
        const float* wcol = Wo + (size_t)(h * C_) * C_ + c; // Wo[h*64+cc, c]
        float acc = 0.0f;
        for (int cc = 0; cc < C_; ++cc) acc = fmaf(vrow[cc], wcol[(size_t)cc * C_], acc);
        vows[(size_t)b * (NH_ * C_) + i] = acc;
    }
}

// ---------------------------------------------------------------------------
// Kernel 2: per wave, 16 rows of q.
//   GEMM1 (WMMA f32 16x16x4): dots[16][48] = q[16][64] @ Kmat^T   (48 wmma)
//   softmax over m=6 per head (LDS staged, v_exp_f32)
//   GEMM2 (WMMA f32 16x16x4): out[16][64] = attn[16][48] @ Vo     (48 wmma)
//   epilogue: + bo + flat residual x, coalesced store.
// ---------------------------------------------------------------------------
__global__ __launch_bounds__(128)
void attn_main_kernel(const float* __restrict__ x, const float* __restrict__ kmat,
                      const float* __restrict__ vo, const float* __restrict__ bo,
                      float* __restrict__ out)
{
    __shared__ float sm[4][16 * NH_];   // 12 KB, one 16x48 strip per wave

    const int lane = threadIdx.x & 31;
    const int wave = threadIdx.x >> 5;
    const int b    = blockIdx.x / 49;
    const int n0   = (blockIdx.x % 49) * 64 + wave * 16;

    const int half = lane >> 4;   // 0: lanes 0-15, 1: lanes 16-31
    const int l16  = lane & 15;

    const float* xb = x    + (size_t)b * (C_ * HW_);
    const float* kb = kmat + (size_t)b * (NH_ * C_);
    const float* vb = vo   + (size_t)b * (NH_ * C_);
    float*      myds = sm[wave];

    // ---------------- GEMM1: dots = q @ Kmat^T (scale pre-folded) ----------
    const v8f z8 = {0.f,0.f,0.f,0.f,0.f,0.f,0.f,0.f};
    v8f acc0 = z8, acc1 = z8, acc2 = z8;
    #pragma unroll 4
    for (int k0 = 0; k0 < C_; k0 += 4) {
        const int c0 = k0 + 2 * half;
        v2f a;                                   // A[M=l16][K=c0], [K=c0+1]
        a.x = xb[(size_t)c0       * HW_ + n0 + l16];
        a.y = xb[(size_t)(c0 + 1) * HW_ + n0 + l16];
        // B[K=c][N=r] = kmat[r][c]; contiguous 8B loads
        const v2f b0 = *(const v2f*)(kb + ( 0 + l16) * C_ + c0);
        const v2f b1 = *(const v2f*)(kb + (16 + l16) * C_ + c0);
        const v2f b2 = *(const v2f*)(kb + (32 + l16) * C_ + c0);
        acc0 = wmma_f32(a, b0, acc0);
        acc1 = wmma_f32(a, b1, acc1);
        acc2 = wmma_f32(a, b2, acc2);
    }
    // Spill D tiles to LDS: lane holds D[M = half*8+i][N = l16] in vgpr i
    #pragma unroll
    for (int i = 0; i < 8; ++i) {
        const int row = half * 8 + i;
        myds[row * NH_ +  0 + l16] = acc0[i];
        myds[row * NH_ + 16 + l16] = acc1[i];
        myds[row * NH_ + 32 + l16] = acc2[i];
    }
    __syncthreads();

    // ---------------- softmax over m=6 within each head ---------------------
    {
        float* p = myds + l16 * NH_ + half * 24;   // each lane: 1 row, 4 heads
        #pragma unroll
        for (int hh = 0; hh < 4; ++hh) {
            float* g = p + hh * M_;
            float mx = g[0];
            #pragma unroll
            for (int t = 1; t < M_; ++t) mx = fmaxf(mx, g[t]);
            float e[M_], s = 0.0f;
            #pragma unroll
            for (int t = 0; t < M_; ++t) { e[t] = __expf(g[t] - mx); s += e[t]; }
            const float inv = __frcp_rn(s);
            #pragma unroll
            for (int t = 0; t < M_; ++t) g[t] = e[t] * inv;
        }
    }
    __syncthreads();

    // ---------------- GEMM2: out = attn @ Vo --------------------------------
    v8f o[4] = {z8, z8, z8, z8};
    #pragma unroll 3
    for (int k0 = 0; k0 < NH_; k0 += 4) {
        const int r0 = k0 + 2 * half;
        const v2f a = *(const v2f*)(myds + l16 * NH_ + r0);   // ds_load_b64
        #pragma unroll
        for (int j = 0; j < 4; ++j) {
            v2f bf;                                // B[K=r][N=c] = Vo[r][c]
            bf.x = vb[(size_t)r0       * C_ + j * 16 + l16];
            bf.y = vb[(size_t)(r0 + 1) * C_ + j * 16 + l16];
            o[j] = wmma_f32(a, bf, o[j]);
        }
    }

    // ---------------- epilogue: + bo + flat residual, coalesced store -------
    float*       ob   = out + (size_t)b * (HW_ * C_);
    const float* xres = x   + (size_t)b * (HW_ * C_);  // residual is flat-aliased
    #pragma unroll
    for (int j = 0; j < 4; ++j) {
        const float bias = bo[j * 16 + l16];
        #pragma unroll
        for (int i = 0; i < 8; ++i) {
            const int    row = n0 + half * 8 + i;
            const size_t idx = (size_t)row * C_ + j * 16 + l16;
            ob[idx] = o[j][i] + bias + xres[idx];
        }
    }
}

// ---------------------------------------------------------------------------
extern "C" void kernel_launch(void* const* d_in, const int* in_sizes, int n_in,
                              void* d_out, int out_size, void* d_ws, size_t ws_size,
                              hipStream_t stream) {
    const float* x  = (const float*)d_in[0];
    const float* z  = (const float*)d_in[1];
    const float* Wk = (const float*)d_in[2];
    const float* bk = (const float*)d_in[3];
    const float* Wv = (const float*)d_in[4];
    const float* bv = (const float*)d_in[5];
    const float* Wo = (const float*)d_in[6];
    const float* bo = (const float*)d_in[7];
    float* out = (float*)d_out;

    // workspace: Kmat [32][48][64] f32, then Vo [32][48][64] f32  (786 KB)
    float* kws  = (float*)d_ws;
    float* vows = kws + (size_t)B_ * NH_ * C_;

    build_kvo_kernel<<<B_, 256, 0, stream>>>(z, Wk, bk, Wv, bv, Wo, kws, vows);
    attn_main_kernel<<<B_ * 49, 128, 0, stream>>>(x, kws, vows, bo, out);
}